// Net_MP_one_68805376082311
// MI455X (gfx1250) — compile-verified
//
#include <hip/hip_runtime.h>
#include <hip/hip_bf16.h>

// ---------------------------------------------------------------------------
// MI455X (gfx1250) implementation of a 4-round GNN message-passing net.
//
// Roofline reasoning:
//  * Node state (N=50000 floats = 200 KB) fits in L2 (192 MB) -> random
//    gathers of h[src] and float atomics into agg[dst] are L2-resident.
//  * HBM traffic ~80 MB (edge_index read 5x, edge_attr once, w_e 5x)
//    -> ~3.5us floor at 23.3 TB/s. Compute is negligible.
//  * The one real matmul (edge MLP, K=3) maps onto V_WMMA_F32_16X16X4_F32
//    (fp32, K padded to 4) so the matrix pipe does it at full precision.
// ---------------------------------------------------------------------------

typedef __attribute__((ext_vector_type(2))) float v2f;
typedef __attribute__((ext_vector_type(8))) float v8f;

// ---------------------------------------------------------------------------
// Edge MLP: w_e[e] = relu(edge_attr[e,0:3] @ w1 + b1) @ w2 + b2
// One wave handles a tile of 16 edges.
//   A (16x4 fp32): lane<16  -> v0=ea[m,0], v1=ea[m,1]
//                  lane>=16 -> v0=ea[m,2], v1=0      (K=3 padded to 4)
//   B (4x16 slice of w1): lane<16 -> v0=w1[0,n], v1=w1[1,n]
//                         lane>=16-> v0=w1[2,n], v1=0
//   C (16x16): every VGPR of lane holds b1[n]  (bias folded into accumulator)
//   D layout: VGPR r = row M=r (lanes 0-15) / M=r+8 (lanes 16-31), col N=lane%16.
// After relu, scale by w2[n] and reduce each row across its 16-lane half.
// ---------------------------------------------------------------------------
__global__ __launch_bounds__(256) void k_edge_mlp(
    const float* __restrict__ edge_attr,   // (E,3)
    const float* __restrict__ w1,          // (3,64) row-major
    const float* __restrict__ b1,          // (64)
    const float* __restrict__ w2,          // (64,1)
    const float* __restrict__ b2,          // (1)
    float* __restrict__ w_e,               // (E) out
    int E)
{
    const int lane = threadIdx.x & 31;
    const int wave = threadIdx.x >> 5;
    const int tile = blockIdx.x * 8 + wave;          // 16 edges per tile
    const int nTiles = (E + 15) >> 4;
    if (tile >= nTiles) return;                       // wave-uniform exit

    const int n0   = lane & 15;
    const int half = lane >> 4;

    // ---- B / C / w2 per N-tile (64 hidden units = 4 tiles of 16) ----
    v2f   bmat[4];
    v8f   cini[4];
    float w2v[4];
#pragma unroll
    for (int t = 0; t < 4; ++t) {
        const int n = n0 + 16 * t;
        bmat[t][0] = half ? w1[2 * 64 + n] : w1[0 * 64 + n];
        bmat[t][1] = half ? 0.0f           : w1[1 * 64 + n];
        const float bb = b1[n];
#pragma unroll
        for (int r = 0; r < 8; ++r) cini[t][r] = bb;
        w2v[t] = w2[n];
    }

    // ---- A: 16 edges x 4 features (feature 3 = 0 pad) ----
    const int m = tile * 16 + n0;
    v2f a;
    if (m < E) {
        a[0] = edge_attr[m * 3 + (half ? 2 : 0)];
        a[1] = half ? 0.0f : edge_attr[m * 3 + 1];
    } else {
        a[0] = 0.0f; a[1] = 0.0f;
    }

    // ---- 4x v_wmma_f32_16x16x4_f32, relu, scale by w2 ----
    float acc[8];
#pragma unroll
    for (int r = 0; r < 8; ++r) acc[r] = 0.0f;

#pragma unroll
    for (int t = 0; t < 4; ++t) {
        v8f d = __builtin_amdgcn_wmma_f32_16x16x4_f32(
            /*neg_a=*/false, a, /*neg_b=*/false, bmat[t],
            /*c_mod=*/(short)0, cini[t], /*reuse_a=*/false, /*reuse_b=*/false);
#pragma unroll
        for (int r = 0; r < 8; ++r) {
            float v = d[r];
            v = v > 0.0f ? v : 0.0f;                  // relu(hidden)
            acc[r] += v * w2v[t];                     // @ w2
        }
    }

    // ---- reduce each row across its 16-lane half (xor masks stay in-half) ----
    const float bias2 = b2[0];
#pragma unroll
    for (int r = 0; r < 8; ++r) {
        float v = acc[r];
        v += __shfl_xor(v, 1, 32);
        v += __shfl_xor(v, 2, 32);
        v += __shfl_xor(v, 4, 32);
        v += __shfl_xor(v, 8, 32);
        if (n0 == 0) {
            const int edge = tile * 16 + r + 8 * half;
            if (edge < E) w_e[edge] = v + bias2;
        }
    }
}

// ---------------------------------------------------------------------------
// Node / edge streaming kernels (memory + L2-atomic bound)
// ---------------------------------------------------------------------------
__global__ __launch_bounds__(256) void k_node_init(
    const float* __restrict__ x, float* __restrict__ deg,
    float* __restrict__ h, int N)
{
    const int i = blockIdx.x * blockDim.x + threadIdx.x;
    if (i < N) { deg[i] = 0.0f; h[i] = x[i * 5 + 2]; }
}

__global__ __launch_bounds__(256) void k_degree(
    const int* __restrict__ dst, float* __restrict__ deg, int E)
{
    const int e = blockIdx.x * blockDim.x + threadIdx.x;
    if (e < E) atomicAdd(&deg[dst[e]], 1.0f);         // global_atomic_add_f32
}

__global__ __launch_bounds__(256) void k_invdeg(float* __restrict__ deg, int N)
{
    const int i = blockIdx.x * blockDim.x + threadIdx.x;
    if (i < N) {
        const float d = deg[i];
        deg[i] = d > 0.0f ? 1.0f / d : 0.0f;          // == 1/max(deg,1) for counts
    }
}

__global__ __launch_bounds__(256) void k_zero(float* __restrict__ p, int N)
{
    const int i = blockIdx.x * blockDim.x + threadIdx.x;
    if (i < N) p[i] = 0.0f;
}

__global__ __launch_bounds__(256) void k_scatter(
    const int* __restrict__ src, const int* __restrict__ dst,
    const float* __restrict__ w_e, const float* __restrict__ h,
    float* __restrict__ agg, int E)
{
    const int e = blockIdx.x * blockDim.x + threadIdx.x;
    if (e < E) {
        const float msg = h[src[e]] * w_e[e];         // h gather: L2-resident
        atomicAdd(&agg[dst[e]], msg);                 // scatter-add into L2
    }
}

__global__ __launch_bounds__(256) void k_update(
    const float* __restrict__ agg, const float* __restrict__ invdeg,
    const float* __restrict__ h_in, const float* __restrict__ root,
    const float* __restrict__ bias, float* __restrict__ h_out, int N)
{
    const int i = blockIdx.x * blockDim.x + threadIdx.x;
    if (i < N) {
        const float v = agg[i] * invdeg[i] + h_in[i] * root[0] + bias[0];
        h_out[i] = v > 0.0f ? v : 0.0f;
    }
}

// ---------------------------------------------------------------------------
extern "C" void kernel_launch(void* const* d_in, const int* in_sizes, int n_in,
                              void* d_out, int out_size, void* d_ws, size_t ws_size,
                              hipStream_t stream)
{
    const float* x    = (const float*)d_in[0];   // (N,5)
    const int*   ei   = (const int*)  d_in[1];   // (2,E)
    const float* ea   = (const float*)d_in[2];   // (E,3)
    const float* w1   = (const float*)d_in[3];   // (3,64)
    const float* b1   = (const float*)d_in[4];   // (64)
    const float* w2   = (const float*)d_in[5];   // (64,1)
    const float* b2   = (const float*)d_in[6];   // (1)
    const float* root = (const float*)d_in[7];   // (1,1)
    const float* bias = (const float*)d_in[8];   // (1)

    const int N = in_sizes[0] / 5;
    const int E = in_sizes[2] / 3;
    const int* srcIdx = ei;
    const int* dstIdx = ei + E;

    // Workspace layout (floats): w_e[E] | deg/invdeg[N] | agg[N] | h[N]
    float* ws   = (float*)d_ws;
    float* w_e  = ws;
    float* deg  = ws + E;
    float* agg  = ws + E + N;
    float* hbuf = ws + E + 2 * N;
    float* out  = (float*)d_out;

    const int nB = (N + 255) / 256;
    const int eB = (E + 255) / 256;
    const int nTiles    = (E + 15) / 16;
    const int mlpBlocks = (nTiles + 7) / 8;          // 8 waves (tiles) per block

    k_edge_mlp<<<mlpBlocks, 256, 0, stream>>>(ea, w1, b1, w2, b2, w_e, E);
    k_node_init<<<nB, 256, 0, stream>>>(x, deg, hbuf, N);
    k_degree<<<eB, 256, 0, stream>>>(dstIdx, deg, E);
    k_invdeg<<<nB, 256, 0, stream>>>(deg, N);

    for (int it = 0; it < 4; ++it) {
        k_zero<<<nB, 256, 0, stream>>>(agg, N);
        k_scatter<<<eB, 256, 0, stream>>>(srcIdx, dstIdx, w_e, hbuf, agg, E);
        float* ho = (it == 3) ? out : hbuf;          // last round writes d_out
        k_update<<<nB, 256, 0, stream>>>(agg, deg, hbuf, root, bias, ho, N);
    }
}